// LocalSpatialEncoding_541165879356
// MI455X (gfx1250) — compile-verified
//
#include <hip/hip_runtime.h>

typedef float v2f __attribute__((ext_vector_type(2)));
typedef float v8f __attribute__((ext_vector_type(8)));

static constexpr int NB = 2;       // batches
static constexpr int NP = 32768;   // points
static constexpr int NK = 16;      // neighbors
static constexpr int ND = 64;      // channels

__device__ __forceinline__ v8f wmma_f32(v2f a, v2f b, v8f c) {
  // V_WMMA_F32_16X16X4_F32: D = A(16x4) x B(4x16) + C(16x16)
  return __builtin_amdgcn_wmma_f32_16x16x4_f32(false, a, false, b, (short)0, c,
                                               false, false);
}

// Build the 3 A fragments for one point. A is 16x12 (channels 0..9 = encoding,
// 10 = 1.0 for bias, 11 = 0). Lane layout per ISA: lanes 0-15 hold K={4j,4j+1},
// lanes 16-31 hold K={4j+2,4j+3}, row m = lane & 15.
__device__ __forceinline__ void build_a(const float* __restrict__ coords,
                                        const float* __restrict__ dist,
                                        const int* __restrict__ idx,
                                        int b, int n, int m, bool hi,
                                        v2f& a0, v2f& a1, v2f& a2) {
  const float* cb = coords + ((size_t)b * NP + n) * 3;
  float cx = cb[0], cy = cb[1], cz = cb[2];
  int nb = idx[((size_t)b * NP + n) * NK + m];
  const float* pb = coords + ((size_t)b * NP + nb) * 3;
  float nx = pb[0], ny = pb[1], nz = pb[2];
  float dd = dist[((size_t)b * NP + n) * NK + m];
  float dx = cx - nx, dy = cy - ny, dz = cz - nz;
  a0.x = hi ? cz : cx;    a0.y = hi ? nx : cy;     // ch {0,1}|{2,3}
  a1.x = hi ? dx : ny;    a1.y = hi ? dy : nz;     // ch {4,5}|{6,7}
  a2.x = hi ? 1.0f : dz;  a2.y = hi ? 0.0f : dd;   // ch {8,9}|{10,11}
}

// B fragment for (d, j): B[c][d] = conv_w[d][c], column d = lane&15 + 16t,
// rows c = 4j + 2*hi + {0,1}; channel 10 -> bias, 11 -> 0.
__device__ __forceinline__ void build_b(const float* __restrict__ w,
                                        const float* __restrict__ bias,
                                        int d, bool hi, int j, v2f& bb) {
  int c0 = 4 * j + (hi ? 2 : 0);              // even, in {0,2,4,6,8,10}
  bb.x = (c0 <= 8) ? w[d * 10 + c0] : bias[d];
  bb.y = (c0 + 1 <= 9) ? w[d * 10 + c0 + 1] : 0.0f;
}

// ---------------- Pass 0: zero the stats accumulators ----------------
__global__ void lse_zero(float* g) {
  if (threadIdx.x < 128) g[threadIdx.x] = 0.0f;
}

// ---------------- Pass 1: per-channel sum / sumsq of x ----------------
// 512 blocks x 256 thr = 4096 waves; each wave handles 16 points.
__global__ __launch_bounds__(256) void lse_stats(
    const float* __restrict__ coords, const float* __restrict__ dist,
    const int* __restrict__ idx, const float* __restrict__ w,
    const float* __restrict__ bias, float* __restrict__ gstats) {
  __shared__ float ls[128];
  int tid = threadIdx.x;
  if (tid < 128) ls[tid] = 0.0f;
  __syncthreads();

  int lane = tid & 31;
  bool hi = lane >= 16;
  int col = lane & 15;
  int wave = (blockIdx.x * 256 + tid) >> 5;   // 0..4095
  float sum[4] = {0.f, 0.f, 0.f, 0.f};
  float sq[4]  = {0.f, 0.f, 0.f, 0.f};

  for (int it = 0; it < 16; ++it) {
    int p = wave + it * 4096;                 // 0..65535
    int b = p >> 15;
    int n = p & (NP - 1);
    v2f a0, a1, a2;
    build_a(coords, dist, idx, b, n, col, hi, a0, a1, a2);
#pragma unroll
    for (int t = 0; t < 4; ++t) {
      int d = 16 * t + col;
      v2f b0, b1, b2;
      build_b(w, bias, d, hi, 0, b0);
      build_b(w, bias, d, hi, 1, b1);
      build_b(w, bias, d, hi, 2, b2);
      v8f acc = {};
      acc = wmma_f32(a0, b0, acc);
      acc = wmma_f32(a1, b1, acc);
      acc = wmma_f32(a2, b2, acc);
#pragma unroll
      for (int r = 0; r < 8; ++r) {
        float v = acc[r];
        sum[t] += v;
        sq[t] += v * v;
      }
    }
  }
  // lanes l and l+16 hold the same channel set: fold halves
#pragma unroll
  for (int t = 0; t < 4; ++t) {
    sum[t] += __shfl_xor(sum[t], 16);
    sq[t]  += __shfl_xor(sq[t], 16);
  }
  if (!hi) {
#pragma unroll
    for (int t = 0; t < 4; ++t) {
      atomicAdd(&ls[16 * t + col], sum[t]);
      atomicAdd(&ls[64 + 16 * t + col], sq[t]);
    }
  }
  __syncthreads();
  if (tid < 128) atomicAdd(&gstats[tid], ls[tid]);
}

// ---------------- Pass 2: stats -> per-channel scale/shift ----------------
__global__ void lse_finalize(const float* __restrict__ gstats,
                             const float* __restrict__ gamma,
                             const float* __restrict__ beta,
                             float* __restrict__ scale,
                             float* __restrict__ shift) {
  int d = threadIdx.x;
  if (d < ND) {
    const float cnt = (float)NB * NP * NK;  // 1048576
    float mean = gstats[d] / cnt;
    float var = gstats[64 + d] / cnt - mean * mean;
    float sc = gamma[d] * rsqrtf(var + 1e-6f);
    scale[d] = sc;
    shift[d] = beta[d] - mean * sc;
  }
}

// ---------------- Pass 3: recompute x, BN+ReLU, write out ----------------
// 8192 blocks x 256 thr = 65536 waves = one wave per (b, n) point.
__global__ __launch_bounds__(256) void lse_main(
    const float* __restrict__ coords, const float* __restrict__ dist,
    const int* __restrict__ idx, const float* __restrict__ w,
    const float* __restrict__ bias, const float* __restrict__ scale,
    const float* __restrict__ shift, float* __restrict__ out) {
  int tid = threadIdx.x;
  int lane = tid & 31;
  bool hi = lane >= 16;
  int col = lane & 15;
  int wave = (blockIdx.x * 256 + tid) >> 5;   // 0..65535
  int b = wave >> 15;
  int n = wave & (NP - 1);

  v2f a0, a1, a2;
  build_a(coords, dist, idx, b, n, col, hi, a0, a1, a2);

#pragma unroll
  for (int t = 0; t < 4; ++t) {
    int d = 16 * t + col;
    v2f b0, b1, b2;
    build_b(w, bias, d, hi, 0, b0);
    build_b(w, bias, d, hi, 1, b1);
    build_b(w, bias, d, hi, 2, b2);
    v8f acc = {};
    acc = wmma_f32(a0, b0, acc);
    acc = wmma_f32(a1, b1, acc);
    acc = wmma_f32(a2, b2, acc);

    float sc = scale[d], sh = shift[d];
    float4 s0, s1;
    s0.x = fmaxf(acc[0] * sc + sh, 0.0f);
    s0.y = fmaxf(acc[1] * sc + sh, 0.0f);
    s0.z = fmaxf(acc[2] * sc + sh, 0.0f);
    s0.w = fmaxf(acc[3] * sc + sh, 0.0f);
    s1.x = fmaxf(acc[4] * sc + sh, 0.0f);
    s1.y = fmaxf(acc[5] * sc + sh, 0.0f);
    s1.z = fmaxf(acc[6] * sc + sh, 0.0f);
    s1.w = fmaxf(acc[7] * sc + sh, 0.0f);
    // out[b][d][n][k], k = 8*hi + r  (8 consecutive floats per lane)
    size_t base = (((size_t)b * 2 * ND + d) * NP + n) * NK + (hi ? 8 : 0);
    *(float4*)(out + base) = s0;
    *(float4*)(out + base + 4) = s1;
  }
}

// ---------------- Feature broadcast: out[b][64+d][n][k] = features[b][d][n] --
__global__ __launch_bounds__(256) void lse_feat(const float* __restrict__ f,
                                                float* __restrict__ out) {
  int p = blockIdx.x * 256 + threadIdx.x;     // 0..4194303 = b*D*N + d*N + n
  float v = f[p];
  int b = p >> 21;           // D*N = 2^21
  int d = (p >> 15) & 63;
  int n = p & (NP - 1);
  size_t base = (((size_t)b * 2 * ND + ND + d) * NP + n) * NK;
  float4 q = make_float4(v, v, v, v);
  float4* o = (float4*)(out + base);
  o[0] = q; o[1] = q; o[2] = q; o[3] = q;     // 16 copies = 64B, coalesced
}

extern "C" void kernel_launch(void* const* d_in, const int* in_sizes, int n_in,
                              void* d_out, int out_size, void* d_ws,
                              size_t ws_size, hipStream_t stream) {
  const float* coords   = (const float*)d_in[0];
  const float* features = (const float*)d_in[1];
  const float* dist     = (const float*)d_in[2];
  const float* conv_w   = (const float*)d_in[3];
  const float* conv_b   = (const float*)d_in[4];
  const float* gamma    = (const float*)d_in[5];
  const float* beta     = (const float*)d_in[6];
  const int*   idx      = (const int*)d_in[7];
  float* out = (float*)d_out;

  float* gstats = (float*)d_ws;   // 128 floats: sum[64], sumsq[64]
  float* scale  = gstats + 128;   // 64
  float* shift  = scale + 64;     // 64

  lse_zero<<<1, 128, 0, stream>>>(gstats);
  lse_stats<<<512, 256, 0, stream>>>(coords, dist, idx, conv_w, conv_b, gstats);
  lse_finalize<<<1, 64, 0, stream>>>(gstats, gamma, beta, scale, shift);
  lse_main<<<8192, 256, 0, stream>>>(coords, dist, idx, conv_w, conv_b, scale,
                                     shift, out);
  lse_feat<<<16384, 256, 0, stream>>>(features, out);
}